// LSTMMRF_20169166422904
// MI455X (gfx1250) — compile-verified
//
#include <hip/hip_runtime.h>
#include <hip/hip_bf16.h>

typedef __attribute__((ext_vector_type(2))) float v2f;
typedef __attribute__((ext_vector_type(8))) float v8f;

#define WAVES 8

// =====================  zero workspace  =====================
__global__ void zero_kernel(float* __restrict__ p, long n) {
    long i = (long)blockIdx.x * blockDim.x + threadIdx.x;
    if (i < n) p[i] = 0.0f;
}

// =====================  weight repack  =====================
// Wp[p][n] = { W[2p][n], W[2p+1][n] }  -> B-fragment pair becomes one b64 load
__global__ void pack_pairs_kernel(const float* __restrict__ src,
                                  float* __restrict__ dst, int K, int Ncols) {
    int i = blockIdx.x * blockDim.x + threadIdx.x;
    int total = (K >> 1) * Ncols;
    if (i < total) {
        int p = i / Ncols, n = i - p * Ncols;
        v2f v;
        v.x = src[(2 * p) * Ncols + n];
        v.y = src[(2 * p + 1) * Ncols + n];
        ((v2f*)dst)[i] = v;
    }
}

// =====================  edge MLP  =====================
// e_inp = [edge_feat | node_feat[src] | node_feat[dst] | g_repr[e2g]]  (128)
// e_out = relu(e_inp @ We1 + be1) @ We2 + be2                          (32)
// side effects: h[dst] += e_out (atomic), ecomb[g] += e_out (LDS->atomic)
__global__ __launch_bounds__(256) void edge_mlp_kernel(
    const float* __restrict__ edge_feat, const float* __restrict__ node_feat,
    const float* __restrict__ g_repr, const int* __restrict__ src,
    const int* __restrict__ dst, const int* __restrict__ e2g,
    const float* __restrict__ We1p, const float* __restrict__ be1,
    const float* __restrict__ We2p, const float* __restrict__ be2,
    float* __restrict__ e_out, float* __restrict__ h_accum,
    float* __restrict__ ecomb, int E, int B)
{
    __shared__ float s_h1[WAVES][16 * 64];
    __shared__ int   s_dst[WAVES][16];
    __shared__ int   s_g[WAVES][16];
    __shared__ float s_ecomb[256];

    const int tid = threadIdx.x;
    s_ecomb[tid] = 0.0f;
    __syncthreads();

    const int wave = tid >> 5;
    const int lane = tid & 31;
    const int m    = lane & 15;
    const int hi   = lane >> 4;    // 0 or 1
    const int koff = hi * 2;

    const int  tileIdx = blockIdx.x * WAVES + wave;
    const long eBase   = (long)tileIdx * 16;

    long e  = eBase + m;
    long ec = (e < E) ? e : (long)(E - 1);
    const int sIdx = src[ec];
    const int dIdx = dst[ec];
    const int gIdx = e2g[ec];
    if (hi == 0) { s_dst[wave][m] = dIdx; s_g[wave][m] = gIdx; }

    const float* seg0 = edge_feat + ec * 32;
    const float* seg1 = node_feat + (long)sIdx * 32;
    const float* seg2 = node_feat + (long)dIdx * 32;
    const float* seg3 = g_repr   + (long)gIdx * 32;

    const v2f* w1 = (const v2f*)We1p;   // [64][64] pairs
    const v2f* w2 = (const v2f*)We2p;   // [32][32] pairs

    // ---------- layer 1: [16x128] @ [128x64] ----------
    v8f acc[4] = {};
    const float* segp[4] = {seg0, seg1, seg2, seg3};
    #pragma unroll
    for (int s = 0; s < 4; ++s) {
        const float* base = segp[s];
        #pragma unroll
        for (int kk = 0; kk < 8; ++kk) {
            const int cbase = 4 * kk + koff;        // col within segment
            const int kglob = s * 32 + cbase;       // global K row (even)
            v2f a = *(const v2f*)(base + cbase);
            const v2f* wp = w1 + (size_t)(kglob >> 1) * 64 + m;
            #pragma unroll
            for (int j = 0; j < 4; ++j) {
                v2f b = wp[j * 16];
                acc[j] = __builtin_amdgcn_wmma_f32_16x16x4_f32(
                    false, a, false, b, (short)0, acc[j], false, false);
            }
        }
    }

    // bias + relu, stage to LDS in row-major [16][64]
    #pragma unroll
    for (int j = 0; j < 4; ++j) {
        const float bias = be1[j * 16 + m];
        #pragma unroll
        for (int r = 0; r < 8; ++r) {
            float v = acc[j][r] + bias;
            v = v > 0.0f ? v : 0.0f;
            s_h1[wave][(r + hi * 8) * 64 + j * 16 + m] = v;
        }
    }
    asm volatile("s_wait_dscnt 0x0" ::: "memory");

    // ---------- layer 2: [16x64] @ [64x32] ----------
    v8f acc2[2] = {};
    const float* hrow = &s_h1[wave][m * 64];
    #pragma unroll
    for (int k = 0; k < 16; ++k) {
        const int cbase = 4 * k + koff;
        v2f a = *(const v2f*)(hrow + cbase);
        const v2f* wp = w2 + (size_t)(cbase >> 1) * 32 + m;
        #pragma unroll
        for (int j = 0; j < 2; ++j) {
            v2f b = wp[j * 16];
            acc2[j] = __builtin_amdgcn_wmma_f32_16x16x4_f32(
                false, a, false, b, (short)0, acc2[j], false, false);
        }
    }

    // bias, store e_out, scatter into h (per-dst) and LDS e_comb (per-graph)
    #pragma unroll
    for (int j = 0; j < 2; ++j) {
        const float bias = be2[j * 16 + m];
        const int   n    = j * 16 + m;
        #pragma unroll
        for (int r = 0; r < 8; ++r) {
            const int  mr   = r + hi * 8;
            const long erow = eBase + mr;
            if (erow < E) {
                const float v = acc2[j][r] + bias;
                e_out[erow * 32 + n] = v;
                unsafeAtomicAdd(&h_accum[(long)s_dst[wave][mr] * 32 + n], v);
                unsafeAtomicAdd(&s_ecomb[s_g[wave][mr] * 32 + n], v);
            }
        }
    }

    __syncthreads();
    if (tid < B * 32) unsafeAtomicAdd(&ecomb[tid], s_ecomb[tid]);
}

// =====================  node MLP  =====================
// n_inp = [node_feat | h | g_repr[n2g]]  (96)
__global__ __launch_bounds__(256) void node_mlp_kernel(
    const float* __restrict__ node_feat, const float* __restrict__ h_accum,
    const float* __restrict__ g_repr, const int* __restrict__ n2g,
    const float* __restrict__ Wn1p, const float* __restrict__ bn1,
    const float* __restrict__ Wn2p, const float* __restrict__ bn2,
    float* __restrict__ n_out, float* __restrict__ ncomb, int N, int B)
{
    __shared__ float s_h1[WAVES][16 * 64];
    __shared__ int   s_g[WAVES][16];
    __shared__ float s_ncomb[256];

    const int tid = threadIdx.x;
    s_ncomb[tid] = 0.0f;
    __syncthreads();

    const int wave = tid >> 5;
    const int lane = tid & 31;
    const int m    = lane & 15;
    const int hi   = lane >> 4;
    const int koff = hi * 2;

    const int  tileIdx = blockIdx.x * WAVES + wave;
    const long nBase   = (long)tileIdx * 16;

    long nd = nBase + m;
    long nc = (nd < N) ? nd : (long)(N - 1);
    const int gIdx = n2g[nc];
    if (hi == 0) s_g[wave][m] = gIdx;

    const float* segp[3] = { node_feat + nc * 32,
                             h_accum   + nc * 32,
                             g_repr    + (long)gIdx * 32 };

    const v2f* w1 = (const v2f*)Wn1p;   // [48][64] pairs
    const v2f* w2 = (const v2f*)Wn2p;   // [32][32] pairs

    // ---------- layer 1: [16x96] @ [96x64] ----------
    v8f acc[4] = {};
    #pragma unroll
    for (int s = 0; s < 3; ++s) {
        const float* base = segp[s];
        #pragma unroll
        for (int kk = 0; kk < 8; ++kk) {
            const int cbase = 4 * kk + koff;
            const int kglob = s * 32 + cbase;
            v2f a = *(const v2f*)(base + cbase);
            const v2f* wp = w1 + (size_t)(kglob >> 1) * 64 + m;
            #pragma unroll
            for (int j = 0; j < 4; ++j) {
                v2f b = wp[j * 16];
                acc[j] = __builtin_amdgcn_wmma_f32_16x16x4_f32(
                    false, a, false, b, (short)0, acc[j], false, false);
            }
        }
    }

    #pragma unroll
    for (int j = 0; j < 4; ++j) {
        const float bias = bn1[j * 16 + m];
        #pragma unroll
        for (int r = 0; r < 8; ++r) {
            float v = acc[j][r] + bias;
            v = v > 0.0f ? v : 0.0f;
            s_h1[wave][(r + hi * 8) * 64 + j * 16 + m] = v;
        }
    }
    asm volatile("s_wait_dscnt 0x0" ::: "memory");

    // ---------- layer 2: [16x64] @ [64x32] ----------
    v8f acc2[2] = {};
    const float* hrow = &s_h1[wave][m * 64];
    #pragma unroll
    for (int k = 0; k < 16; ++k) {
        const int cbase = 4 * k + koff;
        v2f a = *(const v2f*)(hrow + cbase);
        const v2f* wp = w2 + (size_t)(cbase >> 1) * 32 + m;
        #pragma unroll
        for (int j = 0; j < 2; ++j) {
            v2f b = wp[j * 16];
            acc2[j] = __builtin_amdgcn_wmma_f32_16x16x4_f32(
                false, a, false, b, (short)0, acc2[j], false, false);
        }
    }

    #pragma unroll
    for (int j = 0; j < 2; ++j) {
        const float bias = bn2[j * 16 + m];
        const int   n    = j * 16 + m;
        #pragma unroll
        for (int r = 0; r < 8; ++r) {
            const int  mr   = r + hi * 8;
            const long nrow = nBase + mr;
            if (nrow < N) {
                const float v = acc2[j][r] + bias;
                n_out[nrow * 32 + n] = v;
                unsafeAtomicAdd(&s_ncomb[s_g[wave][mr] * 32 + n], v);
            }
        }
    }

    __syncthreads();
    if (tid < B * 32) unsafeAtomicAdd(&ncomb[tid], s_ncomb[tid]);
}

// =====================  global MLP  =====================
// u_inp = [n_comb | e_comb | g_repr] (96), B rows padded to 16. One wave.
__global__ __launch_bounds__(32) void global_mlp_kernel(
    const float* __restrict__ ncomb, const float* __restrict__ ecomb,
    const float* __restrict__ g_repr,
    const float* __restrict__ Wu1p, const float* __restrict__ bu1,
    const float* __restrict__ Wu2p, const float* __restrict__ bu2,
    float* __restrict__ g_out, int B)
{
    __shared__ float s_in[16 * 96];
    __shared__ float s_h1[16 * 64];

    const int lane = threadIdx.x;
    const int m    = lane & 15;
    const int hi   = lane >> 4;
    const int koff = hi * 2;

    for (int idx = lane; idx < 16 * 96; idx += 32) {
        const int row = idx / 96, col = idx % 96;
        float v = 0.0f;
        if (row < B) {
            if (col < 32)      v = ncomb[row * 32 + col];
            else if (col < 64) v = ecomb[row * 32 + (col - 32)];
            else               v = g_repr[row * 32 + (col - 64)];
        }
        s_in[idx] = v;
    }
    asm volatile("s_wait_dscnt 0x0" ::: "memory");

    const v2f* w1 = (const v2f*)Wu1p;   // [48][64] pairs
    const v2f* w2 = (const v2f*)Wu2p;   // [32][32] pairs

    // ---------- layer 1: [16x96] @ [96x64] ----------
    v8f acc[4] = {};
    const float* irow = &s_in[m * 96];
    #pragma unroll
    for (int k = 0; k < 24; ++k) {
        const int cbase = 4 * k + koff;
        v2f a = *(const v2f*)(irow + cbase);
        const v2f* wp = w1 + (size_t)(cbase >> 1) * 64 + m;
        #pragma unroll
        for (int j = 0; j < 4; ++j) {
            v2f b = wp[j * 16];
            acc[j] = __builtin_amdgcn_wmma_f32_16x16x4_f32(
                false, a, false, b, (short)0, acc[j], false, false);
        }
    }

    #pragma unroll
    for (int j = 0; j < 4; ++j) {
        const float bias = bu1[j * 16 + m];
        #pragma unroll
        for (int r = 0; r < 8; ++r) {
            float v = acc[j][r] + bias;
            v = v > 0.0f ? v : 0.0f;
            s_h1[(r + hi * 8) * 64 + j * 16 + m] = v;
        }
    }
    asm volatile("s_wait_dscnt 0x0" ::: "memory");

    // ---------- layer 2: [16x64] @ [64x32] ----------
    v8f acc2[2] = {};
    const float* hrow = &s_h1[m * 64];
    #pragma unroll
    for (int k = 0; k < 16; ++k) {
        const int cbase = 4 * k + koff;
        v2f a = *(const v2f*)(hrow + cbase);
        const v2f* wp = w2 + (size_t)(cbase >> 1) * 32 + m;
        #pragma unroll
        for (int j = 0; j < 2; ++j) {
            v2f b = wp[j * 16];
            acc2[j] = __builtin_amdgcn_wmma_f32_16x16x4_f32(
                false, a, false, b, (short)0, acc2[j], false, false);
        }
    }

    #pragma unroll
    for (int j = 0; j < 2; ++j) {
        const float bias = bu2[j * 16 + m];
        const int   n    = j * 16 + m;
        #pragma unroll
        for (int r = 0; r < 8; ++r) {
            const int mr = r + hi * 8;
            if (mr < B) g_out[mr * 32 + n] = acc2[j][r] + bias;
        }
    }
}

// =====================  launch  =====================
extern "C" void kernel_launch(void* const* d_in, const int* in_sizes, int n_in,
                              void* d_out, int out_size, void* d_ws, size_t ws_size,
                              hipStream_t stream)
{
    const float* edge_feat = (const float*)d_in[0];
    const float* node_feat = (const float*)d_in[1];
    const float* g_repr    = (const float*)d_in[2];
    const int*   src       = (const int*)d_in[3];
    const int*   dst       = (const int*)d_in[4];
    const int*   n2g       = (const int*)d_in[5];
    const int*   e2g       = (const int*)d_in[6];
    const float* We1 = (const float*)d_in[7];  const float* be1 = (const float*)d_in[8];
    const float* We2 = (const float*)d_in[9];  const float* be2 = (const float*)d_in[10];
    const float* Wn1 = (const float*)d_in[11]; const float* bn1 = (const float*)d_in[12];
    const float* Wn2 = (const float*)d_in[13]; const float* bn2 = (const float*)d_in[14];
    const float* Wu1 = (const float*)d_in[15]; const float* bu1 = (const float*)d_in[16];
    const float* Wu2 = (const float*)d_in[17]; const float* bu2 = (const float*)d_in[18];

    const int E = in_sizes[0] / 32;
    const int N = in_sizes[1] / 32;
    const int B = in_sizes[2] / 32;

    float* e_out = (float*)d_out;
    float* n_out = e_out + (size_t)E * 32;
    float* g_out = n_out + (size_t)N * 32;

    // workspace layout
    float* h     = (float*)d_ws;              // N*32
    float* ecomb = h + (size_t)N * 32;        // B*32
    float* ncomb = ecomb + (size_t)B * 32;    // B*32
    float* We1p  = ncomb + (size_t)B * 32;    // 128*64
    float* We2p  = We1p + 128 * 64;           // 64*32
    float* Wn1p  = We2p + 64 * 32;            // 96*64
    float* Wn2p  = Wn1p + 96 * 64;            // 64*32
    float* Wu1p  = Wn2p + 64 * 32;            // 96*64
    float* Wu2p  = Wu1p + 96 * 64;            // 64*32

    const long zn = (long)N * 32 + 2L * B * 32;
    zero_kernel<<<(int)((zn + 255) / 256), 256, 0, stream>>>(h, zn);

    // repack weights into B-fragment pair order
    pack_pairs_kernel<<<(64 * 64 + 255) / 256, 256, 0, stream>>>(We1, We1p, 128, 64);
    pack_pairs_kernel<<<(32 * 32 + 255) / 256, 256, 0, stream>>>(We2, We2p, 64, 32);
    pack_pairs_kernel<<<(48 * 64 + 255) / 256, 256, 0, stream>>>(Wn1, Wn1p, 96, 64);
    pack_pairs_kernel<<<(32 * 32 + 255) / 256, 256, 0, stream>>>(Wn2, Wn2p, 64, 32);
    pack_pairs_kernel<<<(48 * 64 + 255) / 256, 256, 0, stream>>>(Wu1, Wu1p, 96, 64);
    pack_pairs_kernel<<<(32 * 32 + 255) / 256, 256, 0, stream>>>(Wu2, Wu2p, 64, 32);

    const int eTiles = (E + 15) / 16;
    edge_mlp_kernel<<<(eTiles + WAVES - 1) / WAVES, 256, 0, stream>>>(
        edge_feat, node_feat, g_repr, src, dst, e2g,
        We1p, be1, We2p, be2, e_out, h, ecomb, E, B);

    const int nTiles = (N + 15) / 16;
    node_mlp_kernel<<<(nTiles + WAVES - 1) / WAVES, 256, 0, stream>>>(
        node_feat, h, g_repr, n2g,
        Wn1p, bn1, Wn2p, bn2, n_out, ncomb, N, B);

    global_mlp_kernel<<<1, 32, 0, stream>>>(
        ncomb, ecomb, g_repr, Wu1p, bu1, Wu2p, bu2, g_out, B);
}